// Score_39024072852109
// MI455X (gfx1250) — compile-verified
//
#include <hip/hip_runtime.h>

// PHM08 asymmetric score reduction for MI455X (gfx1250, wave32).
// Memory-bound: 160 MB streamed once -> ~6.9 us floor at 23.3 TB/s.

typedef float v4f __attribute__((ext_vector_type(4)));

#define LOG2E_DIV10 0.14426950408889634f   //  log2(e)/10
#define NLOG2E_DIV13 -0.11097654160684334f // -log2(e)/13

__device__ __forceinline__ float score_term(float p, float t) {
    float h = p - t;
    // h >= 0: exp(h/10)-1 ; h < 0: exp(-h/13)-1  -> exp2(h*c)-1 with selected c
    float c = (h >= 0.0f) ? LOG2E_DIV10 : NLOG2E_DIV13;
    return __builtin_amdgcn_exp2f(h * c) - 1.0f;  // v_exp_f32
}

// wave32 butterfly reduction (CDNA5 is wave32-only; width MUST be 32)
__device__ __forceinline__ float wave_reduce(float v) {
#pragma unroll
    for (int off = 16; off > 0; off >>= 1)
        v += __shfl_xor(v, off, 32);
    return v;
}

__global__ __launch_bounds__(256) void phm08_partial(const float* __restrict__ pred,
                                                     const float* __restrict__ gt,
                                                     float* __restrict__ partial,
                                                     int n) {
    const int n4 = n >> 2;
    const v4f* p4 = (const v4f*)pred;
    const v4f* t4 = (const v4f*)gt;

    const int tid    = blockIdx.x * blockDim.x + threadIdx.x;
    const int stride = gridDim.x * blockDim.x;

    // 4 independent accumulator chains for ILP over the exp pipeline
    float a0 = 0.f, a1 = 0.f, a2 = 0.f, a3 = 0.f;

    for (int i = tid; i < n4; i += stride) {
        v4f p = __builtin_nontemporal_load(p4 + i);  // global_load_b128, TH=NT
        v4f t = __builtin_nontemporal_load(t4 + i);
        a0 += score_term(p.x, t.x);
        a1 += score_term(p.y, t.y);
        a2 += score_term(p.z, t.z);
        a3 += score_term(p.w, t.w);
    }
    // tail (n % 4), handled once by global thread 0
    if (tid == 0) {
        for (int i = (n4 << 2); i < n; ++i)
            a0 += score_term(pred[i], gt[i]);
    }

    float acc = (a0 + a1) + (a2 + a3);
    acc = wave_reduce(acc);

    // 256 threads = 8 waves on gfx1250 (wave32)
    __shared__ float lds[8];
    const int lane = threadIdx.x & 31;
    const int wave = threadIdx.x >> 5;
    if (lane == 0) lds[wave] = acc;
    __syncthreads();
    if (wave == 0) {
        float w = (lane < (int)(blockDim.x >> 5)) ? lds[lane] : 0.0f;
        w = wave_reduce(w);
        if (lane == 0) partial[blockIdx.x] = w;
    }
}

// Deterministic final fold of block partials (single block, fixed order).
__global__ __launch_bounds__(256) void phm08_final(const float* __restrict__ partial,
                                                   float* __restrict__ out, int m) {
    float acc = 0.0f;
    for (int i = threadIdx.x; i < m; i += (int)blockDim.x)
        acc += partial[i];
    acc = wave_reduce(acc);

    __shared__ float lds[8];
    const int lane = threadIdx.x & 31;
    const int wave = threadIdx.x >> 5;
    if (lane == 0) lds[wave] = acc;
    __syncthreads();
    if (threadIdx.x == 0) {
        float s = 0.0f;
#pragma unroll
        for (int i = 0; i < 8; ++i) s += lds[i];
        out[0] = s;
    }
}

extern "C" void kernel_launch(void* const* d_in, const int* in_sizes, int n_in,
                              void* d_out, int out_size, void* d_ws, size_t ws_size,
                              hipStream_t stream) {
    const float* pred = (const float*)d_in[0];
    const float* gt   = (const float*)d_in[1];
    float*       out  = (float*)d_out;
    float*       ws   = (float*)d_ws;   // block partials, <= 8 KB

    const int n  = in_sizes[0];
    const int n4 = n >> 2;

    const int BLOCK = 256;
    int grid = (n4 + BLOCK - 1) / BLOCK;
    if (grid > 2048) grid = 2048;   // ~10 float4 iters/thread at N=20M
    if (grid < 1)    grid = 1;

    phm08_partial<<<grid, BLOCK, 0, stream>>>(pred, gt, ws, n);
    phm08_final<<<1, BLOCK, 0, stream>>>(ws, out, grid);
}